// MyBlazeFace_45938970198695
// MI455X (gfx1250) — compile-verified
//
#include <hip/hip_runtime.h>
#include <math.h>

#define NA 896          // anchors
#define NW 28           // wave32s per block (896/32)
#define IOU_THRESH 0.3f
#define MIN_SCORE 0.75f

__global__ __launch_bounds__(NA) void blazeface_wnms_kernel(
    const float* __restrict__ raw_boxes,   // (2048, 896, 16) -> batch 0 used
    const float* __restrict__ raw_scores,  // (2048, 896, 1)  -> batch 0 used
    const float* __restrict__ anchors,     // (896, 4)
    float* __restrict__ out)               // (896, 17)
{
  __shared__ float s_anch[NA * 4];                 // async-staged anchor table
  __shared__ float s_box[4 * NA];                  // decoded [ymin,xmin,ymax,xmax], col-major
  __shared__ float s_wpart[2][NW][18];             // per-wave weighted partials (dbl buf)
  __shared__ float s_bestdet[2][17];               // best row published per step (dbl buf)
  __shared__ unsigned long long s_bestkey[3];      // rotating argmax mailbox

  const int tid  = threadIdx.x;
  const int lane = tid & 31;
  const int wave = tid >> 5;

  // ---- CDNA5 async global->LDS copy of anchors (tracked by ASYNCcnt) ----
  {
    unsigned lds_off = (unsigned)(unsigned long long)(&s_anch[tid * 4]);
    const float* gsrc = anchors + tid * 4;
    asm volatile("global_load_async_to_lds_b128 %0, %1, off"
                 :: "v"(lds_off), "v"(gsrc) : "memory");
  }

  // ---- overlap: this thread's raw box + score while the copy flies ----
  const float4* rb = (const float4*)raw_boxes;     // batch 0
  float4 r0 = rb[tid * 4 + 0];
  float4 r1 = rb[tid * 4 + 1];
  float4 r2 = rb[tid * 4 + 2];
  float4 r3 = rb[tid * 4 + 3];
  float raw_s = raw_scores[tid];                   // batch 0
  raw_s = fminf(fmaxf(raw_s, -100.0f), 100.0f);
  const float score = 1.0f / (1.0f + __expf(-raw_s));

  asm volatile("s_wait_asynccnt 0" ::: "memory");  // own slot is now valid
  const float ax = s_anch[tid * 4 + 0], ay = s_anch[tid * 4 + 1];
  const float aw = s_anch[tid * 4 + 2], ah = s_anch[tid * 4 + 3];

  // ---- decode ----
  const float S = 1.0f / 128.0f;
  float det[17];
  {
    float xc = r0.x * S * aw + ax;
    float yc = r0.y * S * ah + ay;
    float w  = r0.z * S * aw;
    float h  = r0.w * S * ah;
    det[0] = yc - 0.5f * h;  det[1] = xc - 0.5f * w;
    det[2] = yc + 0.5f * h;  det[3] = xc + 0.5f * w;
    det[4]  = r1.x * S * aw + ax;  det[5]  = r1.y * S * ah + ay;
    det[6]  = r1.z * S * aw + ax;  det[7]  = r1.w * S * ah + ay;
    det[8]  = r2.x * S * aw + ax;  det[9]  = r2.y * S * ah + ay;
    det[10] = r2.z * S * aw + ax;  det[11] = r2.w * S * ah + ay;
    det[12] = r3.x * S * aw + ax;  det[13] = r3.y * S * ah + ay;
    det[14] = r3.z * S * aw + ax;  det[15] = r3.w * S * ah + ay;
  }
  det[16] = score;
  const float my_area = (det[2] - det[0]) * (det[3] - det[1]);
  bool remaining = (score >= MIN_SCORE);

  // packed argmax key: higher score wins; equal score -> lower index wins
  const unsigned long long mykey =
      ((unsigned long long)__float_as_uint(score) << 32) |
      (unsigned long long)(0xFFFFFFFFu - (unsigned)tid);

  // read-only decoded box table for the IoU phase
  s_box[0 * NA + tid] = det[0];
  s_box[1 * NA + tid] = det[1];
  s_box[2 * NA + tid] = det[2];
  s_box[3 * NA + tid] = det[3];
  if (tid < 3) s_bestkey[tid] = 0ULL;
  __syncthreads();

  if (remaining) atomicMax(&s_bestkey[0], mykey);  // seed ("step -1" max)
  __syncthreads();

  int count = 0;
  for (int it = 0; it < NA; ++it) {
    const int par = it & 1;

    // ---- loop top: broadcast-read the argmax mailbox ----
    const unsigned long long bk = s_bestkey[it % 3];
    if (bk == 0ULL) break;                         // nothing remaining
    const int best = (int)(0xFFFFFFFFu - (unsigned)(bk & 0xFFFFFFFFull));

    if (tid == best) {                             // publish best row (read post-barrier)
      #pragma unroll
      for (int c = 0; c < 17; ++c) s_bestdet[par][c] = det[c];
    }
    if (tid == 32) s_bestkey[(it + 2) % 3] = 0ULL; // recycle mailbox slot

    // ---- IoU vs best, suppression, next-step argmax atomic ----
    const float by0 = s_box[0 * NA + best], bx0 = s_box[1 * NA + best];
    const float by1 = s_box[2 * NA + best], bx1 = s_box[3 * NA + best];
    const float aA  = (by1 - by0) * (bx1 - bx0);
    float hy = fmaxf(fminf(by1, det[2]) - fmaxf(by0, det[0]), 0.0f);
    float hx = fmaxf(fminf(bx1, det[3]) - fmaxf(bx0, det[1]), 0.0f);
    float inter = hy * hx;
    float iou = inter / (aA + my_area - inter);    // NaN compares false, like jnp
    const bool ov_flag = remaining && (iou > IOU_THRESH);
    remaining = remaining && !ov_flag && (tid != best);
    if (remaining) atomicMax(&s_bestkey[(it + 1) % 3], mykey);

    // ---- weighted partial sums: only waves that contain overlap work ----
    if (__any(ov_flag)) {
      const float wgt  = ov_flag ? score : 0.0f;
      const float oflt = ov_flag ? 1.0f : 0.0f;
      #pragma unroll
      for (int q = 0; q < 18; ++q) {
        float x = (q < 16) ? det[q] * wgt : ((q == 16) ? wgt : oflt);
        #pragma unroll
        for (int off = 16; off > 0; off >>= 1) x += __shfl_down(x, off);
        if (lane == 0) s_wpart[par][wave][q] = x;
      }
    } else if (lane < 18) {
      s_wpart[par][wave][lane] = 0.0f;
    }
    __syncthreads();                               // the single barrier per step

    // ---- wave 0 folds the 28 partials and writes the output row ----
    if (wave == 0) {
      float tot = 0.0f, nf = 0.0f, myq = 0.0f;
      #pragma unroll
      for (int q = 0; q < 18; ++q) {
        float x = (lane < NW) ? s_wpart[par][lane][q] : 0.0f;
        #pragma unroll
        for (int off = 16; off > 0; off >>= 1) x += __shfl_down(x, off);
        x = __shfl(x, 0);
        if (q == 16) tot = x;
        else if (q == 17) nf = x;
        else if (lane == q) myq = x;
      }
      if (lane < 17) {
        float val;
        if (nf > 1.5f) {                           // n > 1 -> weighted merge
          val = (lane < 16) ? myq / (tot > 0.0f ? tot : 1.0f)
                            : tot / fmaxf(nf, 1.0f);
        } else {
          val = s_bestdet[par][lane];              // n <= 1 -> best row verbatim
        }
        out[count * 17 + lane] = val;
      }
    }
    ++count;
  }

  // zero the unused tail of the (896,17) output (d_out is poisoned)
  for (int i = count * 17 + tid; i < NA * 17; i += NA) out[i] = 0.0f;
}

extern "C" void kernel_launch(void* const* d_in, const int* in_sizes, int n_in,
                              void* d_out, int out_size, void* d_ws, size_t ws_size,
                              hipStream_t stream) {
  const float* raw_boxes  = (const float*)d_in[0];
  const float* raw_scores = (const float*)d_in[1];
  const float* anchors    = (const float*)d_in[2];
  float* out = (float*)d_out;
  blazeface_wnms_kernel<<<dim3(1), dim3(NA), 0, stream>>>(
      raw_boxes, raw_scores, anchors, out);
}